// MLEUP_74088185856017
// MI455X (gfx1250) — compile-verified
//
#include <hip/hip_runtime.h>

#define EMB     128
#define KN      32
#define TM      16          // feature rows per block (one WMMA M-tile)
#define XP      260         // padded LDS pitch (floats) to avoid 16-way bank conflicts
#define THREADS 128

typedef __attribute__((ext_vector_type(2))) float v2f;
typedef __attribute__((ext_vector_type(8))) float v8f;

__global__ __launch_bounds__(THREADS)
void gat_fused_kernel(const int*   __restrict__ adj,      // [F, 32]
                      const float* __restrict__ emb_tab,  // [N_NODE, 128]
                      const float* __restrict__ emb_f,    // [F, 128]
                      const float* __restrict__ u,        // [128]
                      const float* __restrict__ W_w,      // [128, 256]
                      const float* __restrict__ W_b,      // [128]
                      float*       __restrict__ out)      // [F, 128]
{
  __shared__ int   s_idx[TM * KN];
  __shared__ float s_u[EMB];
  __shared__ float s_alpha[TM * KN];
  __shared__ float s_X[TM * XP];   // cols [0,128): emb_f, cols [128,256): agg

  const int  tid   = threadIdx.x;            // 0..127
  const long fbase = (long)blockIdx.x * TM;

  // ---- stage u and adjacency tile ----
  if (tid < EMB) s_u[tid] = u[tid];
  #pragma unroll
  for (int i = tid; i < TM * KN; i += THREADS)
    s_idx[i] = adj[fbase * KN + i];
  __syncthreads();

  // ---- phase 1: attention logits (gather is L2-resident: 102MB table < 192MB L2) ----
  #pragma unroll
  for (int p = 0; p < (TM * KN) / THREADS; ++p) {
    const int fk  = tid + p * THREADS;       // flat (f,k) pair
    const int idx = s_idx[fk];
    const float4* row = (const float4*)(emb_tab + (long)idx * EMB);
    __builtin_prefetch(row, 0, 1);           // global_prefetch_b8
    float dot = 0.f;
    #pragma unroll 8
    for (int e = 0; e < EMB / 4; ++e) {
      float4 v = row[e];
      dot = fmaf(v.x, s_u[4*e+0], dot);
      dot = fmaf(v.y, s_u[4*e+1], dot);
      dot = fmaf(v.z, s_u[4*e+2], dot);
      dot = fmaf(v.w, s_u[4*e+3], dot);
    }
    s_alpha[fk] = dot + (idx != 0 ? 0.f : -10000.f);
  }
  __syncthreads();

  // ---- phase 2: softmax over K per f-row (16 rows, 1 thread each) ----
  if (tid < TM) {
    float m = -3.402823466e38f;
    for (int k = 0; k < KN; ++k) m = fmaxf(m, s_alpha[tid * KN + k]);
    float s = 0.f;
    for (int k = 0; k < KN; ++k) s += __expf(s_alpha[tid * KN + k] - m);
    const float inv = 1.f / s;
    for (int k = 0; k < KN; ++k)
      s_alpha[tid * KN + k] = __expf(s_alpha[tid * KN + k] - m) * inv;
  }
  __syncthreads();

  // ---- phase 3: agg = sum_k alpha_k * neigh_k (second L2-hot gather), build X tile ----
  {
    const int f  = tid >> 3;                 // 0..15
    const int ec = (tid & 7) * 16;           // 16-float column chunk
    float acc[16];
    #pragma unroll
    for (int q = 0; q < 16; ++q) acc[q] = 0.f;
    for (int k = 0; k < KN; ++k) {
      const float a = s_alpha[f * KN + k];
      const float4* row =
          (const float4*)(emb_tab + (long)s_idx[f * KN + k] * EMB + ec);
      #pragma unroll
      for (int q = 0; q < 4; ++q) {
        float4 v = row[q];
        acc[4*q+0] = fmaf(a, v.x, acc[4*q+0]);
        acc[4*q+1] = fmaf(a, v.y, acc[4*q+1]);
        acc[4*q+2] = fmaf(a, v.z, acc[4*q+2]);
        acc[4*q+3] = fmaf(a, v.w, acc[4*q+3]);
      }
    }
    float* xrow = &s_X[f * XP];
    #pragma unroll
    for (int q = 0; q < 16; ++q) xrow[EMB + ec + q] = acc[q];
    const float4* ef = (const float4*)(emb_f + (fbase + f) * EMB + ec);
    float4* xd = (float4*)(xrow + ec);
    #pragma unroll
    for (int q = 0; q < 4; ++q) xd[q] = ef[q];
  }
  __syncthreads();

  // ---- phase 4: gate GEMM via V_WMMA_F32_16X16X4_F32 ----
  // Each wave computes output columns [32w, 32w+32): two 16x16 N-tiles, K=256.
  const int lane = tid & 31;
  const int wave = tid >> 5;
  const int l16  = lane & 15;
  const int hi   = lane >> 4;                // lanes 16-31 hold K+2,K+3 / M+8
  const int jb   = wave * 32;

  v8f c0 = {}; v8f c1 = {};
  const float* b0base = W_w + (long)(jb +      l16) * (2 * EMB);
  const float* b1base = W_w + (long)(jb + 16 + l16) * (2 * EMB);
  const float* arow   = &s_X[l16 * XP];

  #pragma unroll 4
  for (int k0 = 0; k0 < 2 * EMB; k0 += 4) {
    const int ks = k0 + 2 * hi;
    v2f a, b0, b1;
    a.x  = arow[ks];     a.y  = arow[ks + 1];   // A frag: 16x4 f32
    b0.x = b0base[ks];   b0.y = b0base[ks + 1]; // B frag: 4x16 (W_w^T), L2-hot
    b1.x = b1base[ks];   b1.y = b1base[ks + 1];
    c0 = __builtin_amdgcn_wmma_f32_16x16x4_f32(false, a, false, b0,
                                               (short)0, c0, false, false);
    c1 = __builtin_amdgcn_wmma_f32_16x16x4_f32(false, a, false, b1,
                                               (short)0, c1, false, false);
  }

  // ---- phase 5: bias + sigmoid gate + blend + store ----
  const float bias0 = W_b[jb + l16];
  const float bias1 = W_b[jb + 16 + l16];
  #pragma unroll
  for (int r = 0; r < 8; ++r) {
    const int  m  = r + 8 * hi;              // C/D layout: VGPR r -> M=r (lanes 0-15), M=r+8 (16-31)
    const long fg = fbase + m;
    {
      const int   j  = jb + l16;
      const float z  = c0[r] + bias0;
      const float g  = 1.f / (1.f + __expf(-z));
      const float e  = s_X[m * XP + j];
      const float ag = s_X[m * XP + EMB + j];
      out[fg * EMB + j] = fmaf(g, e - ag, ag);     // g*e + (1-g)*ag
    }
    {
      const int   j  = jb + 16 + l16;
      const float z  = c1[r] + bias1;
      const float g  = 1.f / (1.f + __expf(-z));
      const float e  = s_X[m * XP + j];
      const float ag = s_X[m * XP + EMB + j];
      out[fg * EMB + j] = fmaf(g, e - ag, ag);
    }
  }
}

extern "C" void kernel_launch(void* const* d_in, const int* in_sizes, int n_in,
                              void* d_out, int out_size, void* d_ws, size_t ws_size,
                              hipStream_t stream) {
  const int*   adj   = (const int*)d_in[0];    // adjacency_fi [F,32] int32
  const float* emb_t = (const float*)d_in[1];  // embedding_i  [N,128]
  const float* embf  = (const float*)d_in[2];  // emb_f        [F,128]
  const float* u     = (const float*)d_in[3];  // u            [128,1]
  const float* W_w   = (const float*)d_in[4];  // W_w          [128,256]
  const float* W_b   = (const float*)d_in[5];  // W_b          [128]
  float* out = (float*)d_out;

  const int F    = in_sizes[0] / KN;           // 50000
  const int grid = (F + TM - 1) / TM;          // 3125 (F divisible by 16)
  gat_fused_kernel<<<grid, THREADS, 0, stream>>>(adj, emb_t, embf, u, W_w, W_b, out);
}